// HGCN_30124900614683
// MI455X (gfx1250) — compile-verified
//
#include <hip/hip_runtime.h>
#include <math.h>

// ---------------------------------------------------------------------------
// HGCN (Poincare ball, 2 layers) for MI455X / gfx1250 (wave32, WMMA).
// GEMM path uses V_WMMA_F32_16X16X4_F32 (native f32 matrix op, D = A*B + C).
// ---------------------------------------------------------------------------

typedef float v2f __attribute__((ext_vector_type(2)));
typedef float v8f __attribute__((ext_vector_type(8)));

#define DD 64
#define TPAD 68           // LDS row stride (odd multiple of 4 -> conflict-free)
#define EPSF 1e-15f
#define ATANH_CLIPF (1.0f - 1e-7f)

// -------------------------------- zero fill --------------------------------
__global__ void k_zero(float* __restrict__ p, long n) {
    long i  = (long)blockIdx.x * blockDim.x + threadIdx.x;
    long st = (long)gridDim.x * blockDim.x;
    for (; i < n; i += st) p[i] = 0.0f;
}

// ------------------------------ degree count -------------------------------
__global__ void k_deg(const int* __restrict__ dst, float* __restrict__ deg, int E) {
    int i = blockIdx.x * blockDim.x + threadIdx.x;
    if (i < E) atomicAdd(&deg[dst[i]], 1.0f);
}

// ---------------- logmap0 + GEMM (h = tangent @ W^T + b) -------------------
// Block = 128 threads (4 waves). Each block handles 64 node rows.
// Each wave computes a 16x64 output tile via 64 x v_wmma_f32_16x16x4_f32.
__global__ __launch_bounds__(128) void k_lin(const float* __restrict__ x,
                                             const float* __restrict__ W,
                                             const float* __restrict__ bias,
                                             float* __restrict__ h,
                                             const float* __restrict__ curv,
                                             int N) {
    __shared__ float sT[DD * TPAD];   // tangent tile, row-major [64][68]
    __shared__ float sW[DD * TPAD];   // W transposed: sW[k][j] = W[j*64 + k]
    __shared__ float sB[DD];
    __shared__ float sScale[DD];

    const int  tid  = threadIdx.x;
    const long base = (long)blockIdx.x * 64;

    const float c  = fabsf(curv[0]);
    const float sc = sqrtf(c);

    // Stage W (transposed) and bias into LDS: 4096 elems / 128 threads.
#pragma unroll
    for (int i = 0; i < 32; ++i) {
        int e = tid + i * 128;
        int j = e >> 6, k = e & 63;
        sW[k * TPAD + j] = W[e];
    }
    if (tid < DD) sB[tid] = bias[tid];

    // Stage 64 x-rows into LDS (coalesced), zero-fill past N.
#pragma unroll
    for (int i = 0; i < 32; ++i) {
        int e = tid + i * 128;
        int r = e >> 6, k = e & 63;
        long gr = base + r;
        sT[r * TPAD + k] = (gr < (long)N) ? x[gr * 64 + k] : 0.0f;
    }
    __syncthreads();

    // logmap0 row scale: atanh(min(sc*||x||, clip)) / (sc*||x||)
    if (tid < DD) {
        float ss = 0.0f;
        const float* row = &sT[tid * TPAD];
#pragma unroll
        for (int k = 0; k < DD; ++k) { float v = row[k]; ss += v * v; }
        float nrm = fmaxf(sqrtf(ss), EPSF);
        float arg = fminf(sc * nrm, ATANH_CLIPF);
        sScale[tid] = atanhf(arg) / (sc * nrm);
    }
    __syncthreads();

#pragma unroll
    for (int i = 0; i < 32; ++i) {
        int e = tid + i * 128;
        int r = e >> 6, k = e & 63;
        sT[r * TPAD + k] *= sScale[r];
    }
    __syncthreads();

    // ------------------------- WMMA section (uniform EXEC) -----------------
    const int lane = tid & 31;
    const int wv   = tid >> 5;        // wave 0..3 -> rows 16*wv .. 16*wv+15
    const int half = lane >> 4;       // 0: K lo pair / M=v ; 1: K hi pair / M=v+8
    const int mr   = lane & 15;       // A: M index; B/D: N index

    v8f acc0 = {0,0,0,0,0,0,0,0};
    v8f acc1 = {0,0,0,0,0,0,0,0};
    v8f acc2 = {0,0,0,0,0,0,0,0};
    v8f acc3 = {0,0,0,0,0,0,0,0};

    const float* arow = &sT[(16 * wv + mr) * TPAD];

#pragma unroll
    for (int s = 0; s < 16; ++s) {
        const int kb = 4 * s + 2 * half;   // this lane's K pair
        // A fragment: 16x4 f32, lane layout per ISA 7.12.2
        v2f a = { arow[kb], arow[kb + 1] };
        // B fragments: 4x16 f32 from transposed-W tile, column group t
        v2f b0 = { sW[kb * TPAD +  0 + mr], sW[(kb + 1) * TPAD +  0 + mr] };
        v2f b1 = { sW[kb * TPAD + 16 + mr], sW[(kb + 1) * TPAD + 16 + mr] };
        v2f b2 = { sW[kb * TPAD + 32 + mr], sW[(kb + 1) * TPAD + 32 + mr] };
        v2f b3 = { sW[kb * TPAD + 48 + mr], sW[(kb + 1) * TPAD + 48 + mr] };
        acc0 = __builtin_amdgcn_wmma_f32_16x16x4_f32(false, a, false, b0, (short)0, acc0, false, false);
        acc1 = __builtin_amdgcn_wmma_f32_16x16x4_f32(false, a, false, b1, (short)0, acc1, false, false);
        acc2 = __builtin_amdgcn_wmma_f32_16x16x4_f32(false, a, false, b2, (short)0, acc2, false, false);
        acc3 = __builtin_amdgcn_wmma_f32_16x16x4_f32(false, a, false, b3, (short)0, acc3, false, false);
    }

    // D layout: VGPR v holds M = v + 8*half, N = mr (per 16x16 f32 C/D table)
    v8f accs[4] = {acc0, acc1, acc2, acc3};
#pragma unroll
    for (int t = 0; t < 4; ++t) {
        const int j  = 16 * t + mr;
        const float bj = sB[j];
#pragma unroll
        for (int v = 0; v < 8; ++v) {
            const int  m  = v + 8 * half;
            const long gr = base + 16 * wv + m;
            if (gr < (long)N) h[gr * 64 + j] = accs[t][v] + bj;
        }
    }
}

// ----------------------- edge scatter: neigh[dst] += h[src] ----------------
// 16 threads per edge, float4 gather + 4 global f32 atomics (coalesced rows).
__global__ void k_scatter(const int* __restrict__ src, const int* __restrict__ dst,
                          const float* __restrict__ h, float* __restrict__ neigh,
                          int E) {
    long tid = (long)blockIdx.x * blockDim.x + threadIdx.x;
    if (tid >= (long)E * 16) return;
    int e = (int)(tid >> 4);
    int q = (int)(tid & 15);
    int s = src[e], d = dst[e];
    const float4 v = ((const float4*)(h + (long)s * 64))[q];
    float* np = neigh + (long)d * 64 + q * 4;
    atomicAdd(np + 0, v.x);
    atomicAdd(np + 1, v.y);
    atomicAdd(np + 2, v.z);
    atomicAdd(np + 3, v.w);
}

// ------------------- inv_deg * neigh, then expmap0 -------------------------
// One wave32 per node row; shuffle reduction for the row norm.
__global__ __launch_bounds__(256) void k_finish(const float* __restrict__ neigh,
                                                const float* __restrict__ deg,
                                                float* __restrict__ out,
                                                const float* __restrict__ curv,
                                                int N) {
    int node = (int)(((long)blockIdx.x * blockDim.x + threadIdx.x) >> 5);
    int lane = threadIdx.x & 31;
    if (node >= N) return;                        // uniform per wave

    const float c  = fabsf(curv[0]);
    const float sc = sqrtf(c);

    float dg  = deg[node];
    float inv = (dg > 0.0f) ? (1.0f / fmaxf(dg, 1.0f)) : 0.0f;

    const float* row = neigh + (long)node * 64;
    float v0 = row[lane]      * inv;
    float v1 = row[lane + 32] * inv;

    float ss = v0 * v0 + v1 * v1;
#pragma unroll
    for (int m = 16; m >= 1; m >>= 1) ss += __shfl_xor(ss, m, 32);

    float nrm   = fmaxf(sqrtf(ss), EPSF);
    float scale = tanhf(sc * nrm) / (sc * nrm);

    float* orow = out + (long)node * 64;
    orow[lane]      = v0 * scale;
    orow[lane + 32] = v1 * scale;
}

// ---------------------------------------------------------------------------
extern "C" void kernel_launch(void* const* d_in, const int* in_sizes, int n_in,
                              void* d_out, int out_size, void* d_ws, size_t ws_size,
                              hipStream_t stream) {
    const int*   src  = (const int*)d_in[0];
    const int*   dst  = (const int*)d_in[1];
    const float* emb  = (const float*)d_in[2];
    const float* W1   = (const float*)d_in[3];
    const float* b1   = (const float*)d_in[4];
    const float* W2   = (const float*)d_in[5];
    const float* b2   = (const float*)d_in[6];
    const float* curv = (const float*)d_in[7];

    const int E = in_sizes[0];
    const int N = in_sizes[2] / DD;
    float* out = (float*)d_out;

    // workspace carve-up: deg[N] | h[N*64] | neigh[N*64]
    float* deg   = (float*)d_ws;
    float* h     = deg + N;
    float* neigh = h + (size_t)N * DD;

    const int  linBlocks  = (N + 63) / 64;
    const long scThreads  = (long)E * 16;
    const int  scBlocks   = (int)((scThreads + 255) / 256);
    const int  finBlocks  = (N + 7) / 8;   // 8 waves of 32 per 256-thread block

    // degree (once)
    k_zero<<<1024, 256, 0, stream>>>(deg, (long)N);
    k_deg<<<(E + 255) / 256, 256, 0, stream>>>(dst, deg, E);

    // ---- layer 1: emb -> out ----
    k_lin<<<linBlocks, 128, 0, stream>>>(emb, W1, b1, h, curv, N);
    k_zero<<<4096, 256, 0, stream>>>(neigh, (long)N * DD);
    k_scatter<<<scBlocks, 256, 0, stream>>>(src, dst, h, neigh, E);
    k_finish<<<finBlocks, 256, 0, stream>>>(neigh, deg, out, curv, N);

    // ---- layer 2: out -> out ----
    k_lin<<<linBlocks, 128, 0, stream>>>(out, W2, b2, h, curv, N);
    k_zero<<<4096, 256, 0, stream>>>(neigh, (long)N * DD);
    k_scatter<<<scBlocks, 256, 0, stream>>>(src, dst, h, neigh, E);
    k_finish<<<finBlocks, 256, 0, stream>>>(neigh, deg, out, curv, N);
}